// EnemyAttention_39565238731277
// MI455X (gfx1250) — compile-verified
//
#include <hip/hip_runtime.h>
#include <math.h>

// MI455X / gfx1250: wave32, WMMA (V_WMMA_F32_16X16X4_F32), 320KB LDS/WGP.
// One wave32 workgroup per batch. HBM-bound (~256MB slots @ 23.3TB/s ~= 11us floor).

typedef __attribute__((ext_vector_type(2))) float v2f;
typedef __attribute__((ext_vector_type(8))) float v8f;

#define PITCH 130           // LDS row pitch in floats (even -> 8B-aligned b64, conflict-free)
#define EPSF 1e-5f
#define NEGINF -3.402823466e38f  // jnp.finfo(float32).min

static __device__ __forceinline__ v8f wmma_f32(v2f a, v2f b, v8f c) {
    return __builtin_amdgcn_wmma_f32_16x16x4_f32(false, a, false, b, (short)0, c, false, false);
}

// ---- Pre-kernel: cls-token q0/k0/v0 are batch-invariant; compute once into d_ws ----
__global__ __launch_bounds__(128)
void cls_qkv_kernel(const float* __restrict__ cls, const float* __restrict__ W_in,
                    const float* __restrict__ b_in, float* __restrict__ qkv0) {
    const int e = threadIdx.x;  // 0..127
    float aq = b_in[e], ak = b_in[128 + e], av = b_in[256 + e];
    const float* wq = W_in + (size_t)e * 128;
    const float* wk = W_in + (size_t)(128 + e) * 128;
    const float* wv = W_in + (size_t)(256 + e) * 128;
    for (int f = 0; f < 128; ++f) {
        const float c = cls[f];
        aq += c * wq[f]; ak += c * wk[f]; av += c * wv[f];
    }
    qkv0[e] = aq; qkv0[128 + e] = ak; qkv0[256 + e] = av;
}

__global__ __launch_bounds__(32)
void enemy_attn_pooled_kernel(const float* __restrict__ slots,     // (B,32,128)
                              const int*   __restrict__ mask,      // (B,32) bool->int
                              const float* __restrict__ W_embed,   // (128,128)
                              const float* __restrict__ b_embed,   // (128,)
                              const float* __restrict__ gamma,     // (128,)
                              const float* __restrict__ beta,      // (128,)
                              const float* __restrict__ W_in,      // (384,128)
                              const float* __restrict__ b_in,      // (384,)
                              const float* __restrict__ W_out,     // (128,128)
                              const float* __restrict__ b_out,     // (128,)
                              const float* __restrict__ qkv0,      // (384,) precomputed
                              float* __restrict__ out)             // (B,128)
{
    __shared__ float s_xk[32 * PITCH];   // x (pre/post LN), then K rows 1..32
    __shared__ float s_v [32 * PITCH];   // V rows 1..32
    __shared__ float s_q0[128];
    __shared__ float s_k0[128];
    __shared__ float s_v0[128];
    __shared__ float s_ctx[128];
    __shared__ float s_w[8 * 32];        // attn weight per head per key-lane
    __shared__ float s_a0[8];            // attn weight of key 0 per head

    const int b      = blockIdx.x;
    const int lane   = threadIdx.x;      // 0..31 (wave32)
    const int l16    = lane & 15;
    const int half   = lane >> 4;        // 0 or 1
    const int koff   = half * 2;         // A/B frag K offset for upper half-wave
    const int rowoff = half * 8;         // C/D frag row offset for upper half-wave

    const float* sb = slots + (size_t)b * (32 * 128);
    const v8f vzero = {0.f, 0.f, 0.f, 0.f, 0.f, 0.f, 0.f, 0.f};

    // ---- Load precomputed cls q0/k0/v0 (hot in L2; 4x b32 per lane per array) ----
    #pragma unroll
    for (int t = 0; t < 4; ++t) {
        const int e = t * 32 + lane;
        s_q0[e] = qkv0[e];
        s_k0[e] = qkv0[128 + e];
        s_v0[e] = qkv0[256 + e];
    }

    // ---- Phase A: x = slots @ W_embed^T + b_embed  (fp32 WMMA 16x16x4) ----
    {
        v8f acc0[8], acc1[8];
        #pragma unroll
        for (int n = 0; n < 8; ++n) { acc0[n] = vzero; acc1[n] = vzero; }

        for (int k = 0; k < 128; k += 4) {
            // A frags: M-tile 0 (rows 0..15) and 1 (rows 16..31); one float2/lane
            const v2f a0 = *(const v2f*)(sb + (size_t)(l16)      * 128 + k + koff);
            const v2f a1 = *(const v2f*)(sb + (size_t)(16 + l16) * 128 + k + koff);
            #pragma unroll
            for (int n = 0; n < 8; ++n) {
                const v2f bb = *(const v2f*)(W_embed + (size_t)(n * 16 + l16) * 128 + k + koff);
                acc0[n] = wmma_f32(a0, bb, acc0[n]);
                acc1[n] = wmma_f32(a1, bb, acc1[n]);
            }
        }
        #pragma unroll
        for (int n = 0; n < 8; ++n) {
            const float be = b_embed[n * 16 + l16];
            #pragma unroll
            for (int v = 0; v < 8; ++v) {
                s_xk[(v + rowoff)      * PITCH + n * 16 + l16] = acc0[n][v] + be;
                s_xk[(16 + v + rowoff) * PITCH + n * 16 + l16] = acc1[n][v] + be;
            }
        }
    }

    // ---- LayerNorm over E=128 (each lane owns one row; v2f LDS traffic) ----
    {
        v2f* row = (v2f*)(s_xk + (size_t)lane * PITCH);   // 8B aligned (PITCH even)
        float mu = 0.f;
        for (int e = 0; e < 64; ++e) { const v2f x = row[e]; mu += x.x + x.y; }
        mu *= (1.f / 128.f);
        float var = 0.f;
        for (int e = 0; e < 64; ++e) {
            const v2f x = row[e];
            const float d0 = x.x - mu, d1 = x.y - mu;
            var += d0 * d0 + d1 * d1;
        }
        var *= (1.f / 128.f);
        const float inv = rsqrtf(var + EPSF);
        for (int e = 0; e < 64; ++e) {
            const v2f x = row[e];
            const v2f g = *(const v2f*)(gamma + e * 2);
            const v2f bt = *(const v2f*)(beta + e * 2);
            v2f y;
            y.x = (x.x - mu) * inv * g.x + bt.x;
            y.y = (x.y - mu) * inv * g.y + bt.y;
            row[e] = y;
        }
    }

    // ---- Phase B: K,V = x_ln @ W_{k,v}^T + b  (fused K/V accumulation) ----
    for (int m = 0; m < 2; ++m) {
        v8f ak[8], av[8];
        #pragma unroll
        for (int n = 0; n < 8; ++n) { ak[n] = vzero; av[n] = vzero; }

        for (int k = 0; k < 128; k += 4) {
            const v2f a = *(const v2f*)(s_xk + (size_t)(m * 16 + l16) * PITCH + k + koff);
            #pragma unroll
            for (int n = 0; n < 8; ++n) {
                const v2f bk = *(const v2f*)(W_in + (size_t)(128 + n * 16 + l16) * 128 + k + koff);
                ak[n] = wmma_f32(a, bk, ak[n]);
                const v2f bv = *(const v2f*)(W_in + (size_t)(256 + n * 16 + l16) * 128 + k + koff);
                av[n] = wmma_f32(a, bv, av[n]);
            }
        }
        // K overwrites x rows of this M-tile (A-frags for this tile are consumed).
        #pragma unroll
        for (int n = 0; n < 8; ++n) {
            const float bkb = b_in[128 + n * 16 + l16];
            const float bvb = b_in[256 + n * 16 + l16];
            #pragma unroll
            for (int v = 0; v < 8; ++v) {
                const int r = m * 16 + v + rowoff;
                s_xk[r * PITCH + n * 16 + l16] = ak[n][v] + bkb;
                s_v [r * PITCH + n * 16 + l16] = av[n][v] + bvb;
            }
        }
    }

    // ---- Attention for the cls query only: lane j <-> key (j+1); key 0 uniform ----
    const int mb = mask[(size_t)b * 32 + lane];  // kpm[j+1] = mask[b][j]
    #pragma unroll
    for (int h = 0; h < 8; ++h) {
        float s0 = 0.f, sj = 0.f;
        #pragma unroll
        for (int d = 0; d < 16; ++d) {
            const float q = s_q0[h * 16 + d];
            s0 += q * s_k0[h * 16 + d];
            sj += q * s_xk[(size_t)lane * PITCH + h * 16 + d];
        }
        s0 *= 0.25f;   // 1/sqrt(DH=16)
        sj *= 0.25f;
        if (mb) sj = NEGINF;

        float mx = sj;
        for (int off = 16; off > 0; off >>= 1) mx = fmaxf(mx, __shfl_xor(mx, off, 32));
        mx = fmaxf(mx, s0);

        const float ej = expf(sj - mx);
        float Z = ej;
        for (int off = 16; off > 0; off >>= 1) Z += __shfl_xor(Z, off, 32);
        const float e0 = expf(s0 - mx);
        Z += e0;

        s_w[h * 32 + lane] = ej / Z;
        if (lane == 0) s_a0[h] = e0 / Z;
    }

    // ---- ctx = attn @ V  (128 outputs over 32 lanes) ----
    for (int t = 0; t < 4; ++t) {
        const int e = t * 32 + lane;
        const int h = e >> 4;
        float c = s_a0[h] * s_v0[e];
        for (int j = 0; j < 32; ++j)
            c += s_w[h * 32 + j] * s_v[(size_t)j * PITCH + e];
        s_ctx[e] = c;
    }

    // ---- pooled = ctx @ W_out^T + b_out  (float4-vectorized dot) ----
    for (int t = 0; t < 4; ++t) {
        const int o = t * 32 + lane;
        float acc = b_out[o];
        const float4* wr = (const float4*)(W_out + (size_t)o * 128);  // 512B-aligned rows
        const float4* cx = (const float4*)s_ctx;
        #pragma unroll 4
        for (int e4 = 0; e4 < 32; ++e4) {
            const float4 w = wr[e4];
            const float4 c = cx[e4];
            acc += c.x * w.x + c.y * w.y + c.z * w.z + c.w * w.w;
        }
        out[(size_t)b * 128 + o] = acc;
    }
}

extern "C" void kernel_launch(void* const* d_in, const int* in_sizes, int n_in,
                              void* d_out, int out_size, void* d_ws, size_t ws_size,
                              hipStream_t stream) {
    const float* slots   = (const float*)d_in[0];
    const int*   mask    = (const int*)  d_in[1];
    const float* W_embed = (const float*)d_in[2];
    const float* b_embed = (const float*)d_in[3];
    const float* gamma   = (const float*)d_in[4];
    const float* beta    = (const float*)d_in[5];
    const float* cls     = (const float*)d_in[6];
    const float* W_in    = (const float*)d_in[7];
    const float* b_in    = (const float*)d_in[8];
    const float* W_out   = (const float*)d_in[9];
    const float* b_out   = (const float*)d_in[10];
    float* out  = (float*)d_out;
    float* qkv0 = (float*)d_ws;   // 384 floats of scratch

    const int B = in_sizes[0] / (32 * 128);  // 16384

    cls_qkv_kernel<<<dim3(1), dim3(128), 0, stream>>>(cls, W_in, b_in, qkv0);
    enemy_attn_pooled_kernel<<<dim3(B), dim3(32), 0, stream>>>(
        slots, mask, W_embed, b_embed, gamma, beta,
        W_in, b_in, W_out, b_out, qkv0, out);
}